// CLML_90099823935622
// MI455X (gfx1250) — compile-verified
//
#include <hip/hip_runtime.h>
#include <hip/hip_bf16.h>

typedef __attribute__((ext_vector_type(2))) float v2f;
typedef __attribute__((ext_vector_type(8))) float v8f;

#define NN   8192
#define CC   64
#define DD   256
#define NCLS 65
#define DSQ  (DD*DD)
#define THRESH 0.84729786f   // ln(0.7/0.3)
#define NS_ITERS 20
#define KS   32              // K-chunk staged in LDS per double-buffer slot

// ---------------------------------------------------------------------------
// 1) membership mask per class (+ pseudo-class 64 = all rows)
//    member = (t==1) || (epoch>=1 && logit > ln(0.7/0.3))
// ---------------------------------------------------------------------------
__global__ __launch_bounds__(256) void mask_kernel(
    const float* __restrict__ logits, const int* __restrict__ targets,
    const int* __restrict__ epoch, float* __restrict__ mask) {
  int idx = blockIdx.x * blockDim.x + threadIdx.x;
  if (idx >= NN * CC) return;
  int row = idx / CC, col = idx % CC;
  bool relabel = (*epoch >= 1);
  bool member = (targets[idx] == 1) || (relabel && logits[idx] > THRESH);
  mask[col * NN + row] = member ? 1.0f : 0.0f;
  if (col == 0) mask[64 * NN + row] = 1.0f;   // "all rows" class
}

// ---------------------------------------------------------------------------
// Async-stage one K-chunk of the A-stripe (cols [bi0,bi0+64)) and B-stripe
// (cols [bj0,bj0+64)) of F into LDS. 128 threads x 8 x b128 = 16KB/chunk,
// tracked on ASYNCcnt (exactly 8 issues per thread -> per-wave count known).
// ---------------------------------------------------------------------------
__device__ __forceinline__ void stage_chunk(
    const float* __restrict__ F, float* ldsA, float* ldsB,
    int k, int bi0, int bj0, int tid) {
  int jj = (tid & 15) * 4;   // 16 groups of 4 floats = 64 cols
  int r0 = tid >> 4;         // 8 base rows per pass
#pragma unroll
  for (int rr = 0; rr < KS; rr += 8) {
    int row = r0 + rr;
    const float* ga = F + (size_t)(k + row) * DD + bi0 + jj;
    const float* gb = F + (size_t)(k + row) * DD + bj0 + jj;
    unsigned la = (unsigned)(unsigned long long)(ldsA + row * 64 + jj);
    unsigned lb = (unsigned)(unsigned long long)(ldsB + row * 64 + jj);
    asm volatile("global_load_async_to_lds_b128 %0, %1, off"
                 :: "v"(la), "v"(ga) : "memory");
    asm volatile("global_load_async_to_lds_b128 %0, %1, off"
                 :: "v"(lb), "v"(gb) : "memory");
  }
}

// ---------------------------------------------------------------------------
// 2) Gram: G_c = F^T diag(m_c) F   (256x256 per class, K = 8192)
//    Block = 4 waves -> 64x64 output; wave -> 32x32 (4 f32-WMMA accums).
//    K-slices double-buffered in LDS via async-to-LDS copies (ASYNCcnt).
// ---------------------------------------------------------------------------
__global__ __launch_bounds__(128) void gram_kernel(
    const float* __restrict__ F, const float* __restrict__ mask,
    float* __restrict__ G) {
  __shared__ float tA[2][KS * 64];
  __shared__ float tB[2][KS * 64];

  int cls  = blockIdx.x >> 4;
  int t    = blockIdx.x & 15;
  int tid  = threadIdx.x;
  int wave = tid >> 5;
  int lane = tid & 31;
  int half = lane >> 4;
  int lm   = lane & 15;
  int bi0 = (t >> 2) * 64;           // block A-col base
  int bj0 = (t & 3)  * 64;           // block B-col base
  int ai  = (wave >> 1) * 32;        // wave offset within A-stripe
  int bjw = (wave & 1)  * 32;        // wave offset within B-stripe
  const float* msk = mask + cls * NN;

  v8f acc00 = {}, acc01 = {}, acc10 = {}, acc11 = {};

  stage_chunk(F, tA[0], tB[0], 0, bi0, bj0, tid);

  for (int c = 0; c < NN / KS; ++c) {
    int k   = c * KS;
    int buf = c & 1;
    if (c + 1 < NN / KS) {
      stage_chunk(F, tA[buf ^ 1], tB[buf ^ 1], k + KS, bi0, bj0, tid);
      asm volatile("s_wait_asynccnt 0x8" ::: "memory");  // chunk c landed (in-order)
    } else {
      asm volatile("s_wait_asynccnt 0x0" ::: "memory");
    }
    __syncthreads();                                      // publish across waves

    const float* la = tA[buf];
    const float* lb = tB[buf];
#pragma unroll
    for (int k4 = 0; k4 < KS; k4 += 4) {
      int ka = k4 + half * 2;        // A VGPR0 = K0|K2, VGPR1 = K1|K3
      int kb = k4 + half;            // B VGPR0 = K0|K1, VGPR1 = K2|K3
      v2f a0, a1, b0, b1;
      a0.x = la[ka * 64 + ai + lm];        a0.y = la[(ka + 1) * 64 + ai + lm];
      a1.x = la[ka * 64 + ai + 16 + lm];   a1.y = la[(ka + 1) * 64 + ai + 16 + lm];
      float w0 = msk[k + kb], w1 = msk[k + kb + 2];
      b0.x = w0 * lb[kb * 64 + bjw + lm];       b0.y = w1 * lb[(kb + 2) * 64 + bjw + lm];
      b1.x = w0 * lb[kb * 64 + bjw + 16 + lm];  b1.y = w1 * lb[(kb + 2) * 64 + bjw + 16 + lm];
      acc00 = __builtin_amdgcn_wmma_f32_16x16x4_f32(false, a0, false, b0, (short)0, acc00, false, false);
      acc01 = __builtin_amdgcn_wmma_f32_16x16x4_f32(false, a0, false, b1, (short)0, acc01, false, false);
      acc10 = __builtin_amdgcn_wmma_f32_16x16x4_f32(false, a1, false, b0, (short)0, acc10, false, false);
      acc11 = __builtin_amdgcn_wmma_f32_16x16x4_f32(false, a1, false, b1, (short)0, acc11, false, false);
    }
    __syncthreads();                 // done reading buf before it is re-staged
  }

  float* g = G + cls * DSQ;
  int bi = bi0 + ai, bj = bj0 + bjw;
#pragma unroll
  for (int r = 0; r < 8; ++r) {      // C/D layout: VGPR r -> M=r (lo lanes) / r+8 (hi)
    int gr0 = bi + r + half * 8;
    int gr1 = gr0 + 16;
    g[gr0 * DD + bj + lm]      = acc00[r];
    g[gr0 * DD + bj + 16 + lm] = acc01[r];
    g[gr1 * DD + bj + lm]      = acc10[r];
    g[gr1 * DD + bj + 16 + lm] = acc11[r];
  }
}

// ---------------------------------------------------------------------------
// 3) Newton-Schulz init: alpha = ||G||_F (>= lambda_max), Y0 = G/alpha, Z0 = I
// ---------------------------------------------------------------------------
__global__ __launch_bounds__(256) void init_kernel(
    const float* __restrict__ G, float* __restrict__ Y, float* __restrict__ Z,
    float* __restrict__ alpha) {
  int cls = blockIdx.x;
  const float* g = G + cls * DSQ;
  __shared__ float red[256];
  __shared__ float a_s;
  float s = 0.f;
  for (int i = threadIdx.x; i < DSQ; i += 256) { float v = g[i]; s += v * v; }
  red[threadIdx.x] = s;
  __syncthreads();
  for (int off = 128; off > 0; off >>= 1) {
    if (threadIdx.x < off) red[threadIdx.x] += red[threadIdx.x + off];
    __syncthreads();
  }
  if (threadIdx.x == 0) {
    a_s = fmaxf(sqrtf(red[0]), 1e-30f);  // guard empty class (G == 0)
    alpha[cls] = a_s;
  }
  __syncthreads();
  float inva = 1.0f / a_s;
  float* y = Y + cls * DSQ;
  float* z = Z + cls * DSQ;
  for (int i = threadIdx.x; i < DSQ; i += 256) {
    y[i] = g[i] * inva;
    z[i] = ((i / DD) == (i % DD)) ? 1.0f : 0.0f;
  }
}

// ---------------------------------------------------------------------------
// 4) 256x256x256 per-class GEMM via f32 WMMA.
//    mode 0: OUT = A*B          mode 1: OUT = 1.5*I - 0.5*(A*B)
//    Block = 4 waves -> 64x64 tile; wave -> 16x64 strip (4 accumulators).
//    Per-class working set (3 x 256KB) lives in L2 (192MB global L2).
// ---------------------------------------------------------------------------
__global__ __launch_bounds__(128) void gemm_kernel(
    const float* __restrict__ A, const float* __restrict__ B,
    float* __restrict__ OUT, int mode) {
  int cls  = blockIdx.y;
  int bx   = blockIdx.x;
  int rb   = (bx >> 2) * 64;
  int cb   = (bx & 3)  * 64;
  int wave = threadIdx.x >> 5;
  int lane = threadIdx.x & 31;
  int half = lane >> 4;
  int lm   = lane & 15;
  int row  = rb + wave * 16 + lm;
  const float* a = A + cls * DSQ;
  const float* b = B + cls * DSQ;

  v8f acc0 = {}, acc1 = {}, acc2 = {}, acc3 = {};

#pragma unroll 4
  for (int k = 0; k < DD; k += 4) {
    v2f av = *(const v2f*)&a[row * DD + k + half * 2];   // K0,K1 | K2,K3
    const float* br0 = &b[(k + half) * DD + cb + lm];    // K0|K1 rows
    const float* br1 = &b[(k + 2 + half) * DD + cb + lm];// K2|K3 rows
    v2f b0, b1, b2, b3;
    b0.x = br0[0];  b0.y = br1[0];
    b1.x = br0[16]; b1.y = br1[16];
    b2.x = br0[32]; b2.y = br1[32];
    b3.x = br0[48]; b3.y = br1[48];
    acc0 = __builtin_amdgcn_wmma_f32_16x16x4_f32(false, av, false, b0, (short)0, acc0, false, false);
    acc1 = __builtin_amdgcn_wmma_f32_16x16x4_f32(false, av, false, b1, (short)0, acc1, false, false);
    acc2 = __builtin_amdgcn_wmma_f32_16x16x4_f32(false, av, false, b2, (short)0, acc2, false, false);
    acc3 = __builtin_amdgcn_wmma_f32_16x16x4_f32(false, av, false, b3, (short)0, acc3, false, false);
  }

  float* o = OUT + cls * DSQ;
#pragma unroll
  for (int r = 0; r < 8; ++r) {
    int gr = rb + wave * 16 + r + half * 8;
    float v0 = acc0[r], v1 = acc1[r], v2 = acc2[r], v3 = acc3[r];
    int c0 = cb + lm, c1 = cb + 16 + lm, c2 = cb + 32 + lm, c3 = cb + 48 + lm;
    if (mode == 1) {   // fused M = 1.5I - 0.5*T epilogue
      v0 = ((gr == c0) ? 1.5f : 0.0f) - 0.5f * v0;
      v1 = ((gr == c1) ? 1.5f : 0.0f) - 0.5f * v1;
      v2 = ((gr == c2) ? 1.5f : 0.0f) - 0.5f * v2;
      v3 = ((gr == c3) ? 1.5f : 0.0f) - 0.5f * v3;
    }
    o[gr * DD + c0] = v0;
    o[gr * DD + c1] = v1;
    o[gr * DD + c2] = v2;
    o[gr * DD + c3] = v3;
  }
}

// ---------------------------------------------------------------------------
// 5) nuc_c = sqrt(alpha_c) * trace(Y_final_c);  scalar objective
// ---------------------------------------------------------------------------
__global__ __launch_bounds__(128) void finalize_kernel(
    const float* __restrict__ Yf, const float* __restrict__ alpha,
    const float* __restrict__ lam, float* __restrict__ out) {
  __shared__ float nuc[NCLS];
  int t = threadIdx.x;
  if (t < NCLS) {
    float tr = 0.f;
    const float* y = Yf + t * DSQ;
    for (int i = 0; i < DD; ++i) tr += y[i * DD + i];
    nuc[t] = sqrtf(alpha[t]) * tr;
  }
  __syncthreads();
  if (t == 0) {
    float obj = 0.f;
    for (int c = 0; c < CC; ++c) obj += fmaxf(nuc[c], 1.0f);
    float l = *lam;
    out[0] = (obj - l * nuc[64]) / (float)NN * l;
  }
}

// ---------------------------------------------------------------------------
extern "C" void kernel_launch(void* const* d_in, const int* in_sizes, int n_in,
                              void* d_out, int out_size, void* d_ws, size_t ws_size,
                              hipStream_t stream) {
  const float* logits  = (const float*)d_in[0];
  const int*   targets = (const int*)  d_in[1];
  const float* feature = (const float*)d_in[2];
  const float* lam     = (const float*)d_in[3];
  const int*   epoch   = (const int*)  d_in[4];

  float* ws   = (float*)d_ws;
  float* mask = ws;                      // 65*8192
  float* G    = mask + (size_t)NCLS*NN;  // 65*65536 ; reused as M after init
  float* Y0   = G  + (size_t)NCLS*DSQ;
  float* Y1   = Y0 + (size_t)NCLS*DSQ;
  float* Z0   = Y1 + (size_t)NCLS*DSQ;
  float* Z1   = Z0 + (size_t)NCLS*DSQ;
  float* alpha= Z1 + (size_t)NCLS*DSQ;   // 65 floats

  mask_kernel<<<(NN*CC + 255)/256, 256, 0, stream>>>(logits, targets, epoch, mask);
  gram_kernel<<<NCLS*16, 128, 0, stream>>>(feature, mask, G);
  init_kernel<<<NCLS, 256, 0, stream>>>(G, Y0, Z0, alpha);

  float* M  = G;   // G no longer needed after init
  float* Yc = Y0; float* Yn = Y1;
  float* Zc = Z0; float* Zn = Z1;
  dim3 grid16(16, NCLS);
  for (int it = 0; it < NS_ITERS; ++it) {
    gemm_kernel<<<grid16, 128, 0, stream>>>(Zc, Yc, M,  1);  // M  = 1.5I - 0.5*Z*Y
    gemm_kernel<<<grid16, 128, 0, stream>>>(Yc, M,  Yn, 0);  // Y' = Y*M
    gemm_kernel<<<grid16, 128, 0, stream>>>(M,  Zc, Zn, 0);  // Z' = M*Z
    float* ty = Yc; Yc = Yn; Yn = ty;
    float* tz = Zc; Zc = Zn; Zn = tz;
  }

  finalize_kernel<<<1, 128, 0, stream>>>(Yc, alpha, lam, (float*)d_out);
}